// SceneStructureBlock_64020782514346
// MI455X (gfx1250) — compile-verified
//
#include <hip/hip_runtime.h>

typedef float v2f __attribute__((ext_vector_type(2)));
typedef float v8f __attribute__((ext_vector_type(8)));

#define DD    512      // in_dims
#define BATCH 64
#define NP    128      // patches
#define NM    1152     // P*PS*PS (mask outputs)
#define NT    768      // 6*P (theta outputs)
#define SS    64       // scene size

// ---------------------------------------------------------------------------
// Kernel 1: fused GEMM (z @ [Wm;Wt]^T) via V_WMMA_F32_16X16X4_F32.
// One wave32 computes one 16x16 output tile; K=512 -> 128 WMMA steps.
// Epilogue applies bias + (1 - sigmoid) * noise for masks, bias for theta.
// ---------------------------------------------------------------------------
__global__ __launch_bounds__(128)
void scene_gemm_wmma(const float* __restrict__ z,
                     const float* __restrict__ Wm, const float* __restrict__ bm,
                     const float* __restrict__ Wt, const float* __restrict__ bt,
                     const float* __restrict__ noise,
                     float* __restrict__ masks, float* __restrict__ theta)
{
    const int wave  = threadIdx.x >> 5;
    const int tile  = blockIdx.x * 4 + wave;   // 480 tiles total
    const int ntile = tile >> 2;               // 0..119 (N direction)
    const int mtile = tile & 3;                // 0..3   (M direction)
    const int lane  = threadIdx.x & 31;
    const int lo    = lane & 15;
    const int hi    = lane >> 4;

    const int n0 = ntile * 16;
    const int m0 = mtile * 16;

    // A: z row for M = m0+lo ; B: weight row for N = n0+lo (B[k,n] = W[n,k]).
    const float* zrow = z + (m0 + lo) * DD;
    const float* wrow = (n0 < NM) ? (Wm + (n0 + lo) * DD)
                                  : (Wt + (n0 - NM + lo) * DD);

    v8f acc = {0.f, 0.f, 0.f, 0.f, 0.f, 0.f, 0.f, 0.f};

#pragma unroll 8
    for (int k = 0; k < DD; k += 4) {
        // 16x4 f32 A layout: lanes 0-15 carry K = k+0,k+1 ; lanes 16-31 K = k+2,k+3
        v2f a = *(const v2f*)(zrow + k + 2 * hi);
        // 4x16 f32 B layout mirrors it (VGPR r = row K=r, halves offset by 2)
        v2f b = *(const v2f*)(wrow + k + 2 * hi);
        acc = __builtin_amdgcn_wmma_f32_16x16x4_f32(
            /*neg_a=*/false, a, /*neg_b=*/false, b,
            /*c_mod=*/(short)0, acc, /*reuse_a=*/false, /*reuse_b=*/false);
    }

    const int col = n0 + lo;
    if (n0 < NM) {
        // masks = (1 - sigmoid(x)) * noise = noise / (1 + exp(x))
        const float nz   = noise[col % 9];
        const float bias = bm[col];
#pragma unroll
        for (int r = 0; r < 8; ++r) {
            const int   row = m0 + r + 8 * hi;      // C/D layout: VGPR r -> M=r(+8)
            const float x   = acc[r] + bias;
            masks[row * NM + col] = nz / (1.0f + __expf(x));
        }
    } else {
        const int   c    = col - NM;
        const float bias = bt[c];
#pragma unroll
        for (int r = 0; r < 8; ++r) {
            const int row = m0 + r + 8 * hi;
            theta[row * NT + c] = acc[r] + bias;
        }
    }
}

// ---------------------------------------------------------------------------
// Kernel 2: affine-grid + bilinear(zero-pad) resample of 3x3 patch -> 64x64.
// Bilinear w/ zero padding over a 3x3 image == separable hat-kernel sum:
//   out = sum_{i,j} img[i,j] * tri(iy - i) * tri(ix - j),  tri(t)=max(0,1-|t|)
// Branch-free, no dynamic indexing. float4 coalesced stores (128 MB total).
// ---------------------------------------------------------------------------
__device__ __forceinline__ float tri_w(float t) {
    return fmaxf(0.0f, 1.0f - fabsf(t));
}

__global__ __launch_bounds__(256)
void scene_sample(const float* __restrict__ masks,
                  const float* __restrict__ theta,
                  float* __restrict__ out)
{
    __shared__ float simg[9];
    __shared__ float sth[6];

    const int pid = blockIdx.x;        // b*128 + p
    const int b   = pid >> 7;
    const int p   = pid & 127;
    const int tid = threadIdx.x;

    if (tid < 9)                 simg[tid]      = masks[b * NM + p * 9 + tid];
    if (tid >= 16 && tid < 22)   sth[tid - 16]  = theta[b * NT + p * 6 + (tid - 16)];
    __syncthreads();

    const float t0 = sth[0], t1 = sth[1], t2 = sth[2];
    const float t3 = sth[3], t4 = sth[4], t5 = sth[5];
    float img[9];
#pragma unroll
    for (int i = 0; i < 9; ++i) img[i] = simg[i];

    float4* out4 = (float4*)(out + (size_t)pid * (SS * SS));

#pragma unroll
    for (int k = 0; k < 4; ++k) {
        const int q    = k * 256 + tid;       // quad index 0..1023
        const int base = q * 4;               // first pixel of the quad
        const int h    = base >> 6;
        const int w0   = base & 63;
        const float cy = (h + 0.5f) * (2.0f / SS) - 1.0f;

        float rv[4];
#pragma unroll
        for (int j = 0; j < 4; ++j) {
            const float cx = (w0 + j + 0.5f) * (2.0f / SS) - 1.0f;
            const float gx = t0 * cx + t1 * cy + t2;          // grid x in [-1,1]
            const float gy = t3 * cx + t4 * cy + t5;          // grid y
            const float ix = ((gx + 1.0f) * 3.0f - 1.0f) * 0.5f;  // align_corners=False
            const float iy = ((gy + 1.0f) * 3.0f - 1.0f) * 0.5f;

            const float wx0 = tri_w(ix),        wx1 = tri_w(ix - 1.0f), wx2 = tri_w(ix - 2.0f);
            const float wy0 = tri_w(iy),        wy1 = tri_w(iy - 1.0f), wy2 = tri_w(iy - 2.0f);

            const float r0 = img[0] * wx0 + img[1] * wx1 + img[2] * wx2;
            const float r1 = img[3] * wx0 + img[4] * wx1 + img[5] * wx2;
            const float r2 = img[6] * wx0 + img[7] * wx1 + img[8] * wx2;
            rv[j] = wy0 * r0 + wy1 * r1 + wy2 * r2;
        }
        out4[q] = make_float4(rv[0], rv[1], rv[2], rv[3]);
    }
}

// ---------------------------------------------------------------------------
extern "C" void kernel_launch(void* const* d_in, const int* in_sizes, int n_in,
                              void* d_out, int out_size, void* d_ws, size_t ws_size,
                              hipStream_t stream)
{
    (void)in_sizes; (void)n_in; (void)out_size; (void)ws_size;

    const float* z     = (const float*)d_in[0];
    const float* Wm    = (const float*)d_in[1];
    const float* bm    = (const float*)d_in[2];
    const float* Wt    = (const float*)d_in[3];
    const float* bt    = (const float*)d_in[4];
    const float* noise = (const float*)d_in[5];

    float* masks = (float*)d_ws;              // 64*1152 f32
    float* theta = masks + BATCH * NM;        // 64*768  f32  (total ~0.5 MB)
    float* out   = (float*)d_out;             // 64*128*64*64 f32

    // 480 tiles = 120 blocks x 4 waves (one 16x16 WMMA tile per wave)
    scene_gemm_wmma<<<120, 128, 0, stream>>>(z, Wm, bm, Wt, bt, noise, masks, theta);
    // one block per patch, float4 stores
    scene_sample<<<BATCH * NP, 256, 0, stream>>>(masks, theta, out);
}